// SelectiveSSM_59579786330316
// MI455X (gfx1250) — compile-verified
//
#include <hip/hip_runtime.h>
#include <hip/hip_bf16.h>

#define B_SZ 4
#define L_SZ 2048
#define D_SZ 1024
#define N_SZ 16
#define P_SZ 33   // 2*N + 1 projection columns

typedef float v2f __attribute__((ext_vector_type(2)));
typedef float v8f __attribute__((ext_vector_type(8)));

// ---------------------------------------------------------------------------
// Kernel 1: depthwise causal conv1d (K=4) + bias + SiLU  ->  xc (B,L,D)
// One thread per output element, coalesced along D.
// ---------------------------------------------------------------------------
__global__ __launch_bounds__(256) void conv_silu_kernel(
    const float* __restrict__ x, const float* __restrict__ cw,
    const float* __restrict__ cb, float* __restrict__ xc)
{
  const int idx = blockIdx.x * 256 + threadIdx.x;   // over B*L*D
  const int d  = idx & (D_SZ - 1);
  const int bl = idx >> 10;                         // b*L + l
  const int l  = bl & (L_SZ - 1);
  const float* xp = x + (size_t)bl * D_SZ + d;      // x[b, l, d]
  const float* wp = cw + (size_t)d * 4;
  float acc = cb[d];
#pragma unroll
  for (int j = 0; j < 4; ++j) {
    const int li = l - 3 + j;
    const float xv = (li >= 0) ? xp[(j - 3) * D_SZ] : 0.f;
    acc = fmaf(xv, wp[j], acc);
  }
  xc[idx] = acc / (1.f + __expf(-acc));             // SiLU
}

// ---------------------------------------------------------------------------
// Kernel 2: proj = xc (8192x1024) @ xproj_w^T (1024x33) via fp32 WMMA.
// One wave per 16-row M tile; 3 N-tiles of 16 cover the 33 columns (padding
// zeroed arithmetically so EXEC stays all-ones for the WMMA ops).
// A layout (16x4 f32): vgpr=k&1, lane=(k>=2)*16+m.  B mirrored with n.
// ---------------------------------------------------------------------------
__global__ __launch_bounds__(128) void proj_wmma_kernel(
    const float* __restrict__ xc, const float* __restrict__ w,
    float* __restrict__ proj)
{
  const int wave = blockIdx.x * 4 + (threadIdx.x >> 5);   // m-tile, 0..511
  const int lane = threadIdx.x & 31;
  const int mrow = wave * 16 + (lane & 15);
  const int kb   = (lane >> 4) << 1;                      // 0 or 2
  const float* arow = xc + (size_t)mrow * D_SZ + kb;

  const int ncol = lane & 15;
  const int p0 = ncol;            // tile 0: cols  0..15
  const int p1 = 16 + ncol;       // tile 1: cols 16..31
  int p2 = 32 + ncol;             // tile 2: col 32 + padding
  const float s2 = (p2 < P_SZ) ? 1.f : 0.f;
  if (p2 >= P_SZ) p2 = P_SZ - 1;  // clamp address, zero the data
  const float* brow0 = w + (size_t)p0 * D_SZ + kb;
  const float* brow1 = w + (size_t)p1 * D_SZ + kb;
  const float* brow2 = w + (size_t)p2 * D_SZ + kb;

  v8f acc0 = {}, acc1 = {}, acc2 = {};
  for (int k0 = 0; k0 < D_SZ; k0 += 4) {
    v2f a  = *(const v2f*)(arow  + k0);
    v2f b0 = *(const v2f*)(brow0 + k0);
    v2f b1 = *(const v2f*)(brow1 + k0);
    v2f b2 = *(const v2f*)(brow2 + k0);
    b2 *= s2;
    acc0 = __builtin_amdgcn_wmma_f32_16x16x4_f32(false, a, false, b0,
                                                 (short)0, acc0, false, false);
    acc1 = __builtin_amdgcn_wmma_f32_16x16x4_f32(false, a, false, b1,
                                                 (short)0, acc1, false, false);
    acc2 = __builtin_amdgcn_wmma_f32_16x16x4_f32(false, a, false, b2,
                                                 (short)0, acc2, false, false);
  }

  // D layout: vgpr v -> m = v + 8*(lane>=16), n = lane&15
  const int mhi = (lane >> 4) << 3;
#pragma unroll
  for (int v = 0; v < 8; ++v) {
    const int row = wave * 16 + v + mhi;
    float* pr = proj + (size_t)row * P_SZ;
    pr[p0] = acc0[v];
    pr[p1] = acc1[v];
    if (32 + ncol < P_SZ) pr[32 + ncol] = acc2[v];
  }
}

// ---------------------------------------------------------------------------
// Kernel 3: selective scan. One lane per (b, d, n); 16-lane groups share a
// channel, state h lives in a register, y reduced across n with shfl_xor.
// delta recomputed from proj[...,32] (rank-1 dt projection + softplus).
// ---------------------------------------------------------------------------
__global__ __launch_bounds__(256) void scan_kernel(
    const float* __restrict__ x, const float* __restrict__ xc,
    const float* __restrict__ proj, const float* __restrict__ dt_w,
    const float* __restrict__ dt_b, const float* __restrict__ a_log,
    const float* __restrict__ Dp, float* __restrict__ out)
{
  const int n = threadIdx.x & 15;
  const int c = blockIdx.x * 16 + (threadIdx.x >> 4);  // channel in [0, B*D)
  const int b = c >> 10;
  const int d = c & (D_SZ - 1);

  const float a_c = -__expf(a_log[d * N_SZ + n]);      // -exp(a_log)
  const float dtw = dt_w[d];
  const float dtb = dt_b[d];
  const float dpd = Dp[d];

  const float* pr   = proj + (size_t)b * L_SZ * P_SZ;
  const float* xr   = x    + (size_t)b * L_SZ * D_SZ + d;
  const float* xcr  = xc   + (size_t)b * L_SZ * D_SZ + d;
  float*       outr = out  + (size_t)b * L_SZ * D_SZ + d;

  float h = 0.f;
  for (int t = 0; t < L_SZ; ++t) {
    const float* pt = pr + (size_t)t * P_SZ;
    const float bt  = pt[n];
    const float ct  = pt[N_SZ + n];
    const float p32 = pt[2 * N_SZ];
    const float xt  = xcr[(size_t)t * D_SZ];

    const float z     = fmaf(p32, dtw, dtb);
    const float delta = (z > 20.f) ? z : __logf(1.f + __expf(z));  // softplus
    const float ab    = __expf(delta * a_c);

    h = fmaf(ab, h, delta * xt * bt);
    float y = h * ct;
    y += __shfl_xor(y, 1);
    y += __shfl_xor(y, 2);
    y += __shfl_xor(y, 4);
    y += __shfl_xor(y, 8);
    if (n == 0)
      outr[(size_t)t * D_SZ] = fmaf(xr[(size_t)t * D_SZ], dpd, y);
  }
}

// ---------------------------------------------------------------------------
extern "C" void kernel_launch(void* const* d_in, const int* in_sizes, int n_in,
                              void* d_out, int out_size, void* d_ws, size_t ws_size,
                              hipStream_t stream) {
  const float* x      = (const float*)d_in[0];
  const float* conv_w = (const float*)d_in[1];
  const float* conv_b = (const float*)d_in[2];
  const float* xproj  = (const float*)d_in[3];
  const float* dt_w   = (const float*)d_in[4];
  const float* dt_b   = (const float*)d_in[5];
  const float* a_log  = (const float*)d_in[6];
  const float* Dp     = (const float*)d_in[7];
  float* out = (float*)d_out;

  float* xc   = (float*)d_ws;                                         // 32 MB
  float* proj = (float*)((char*)d_ws +
                         (size_t)B_SZ * L_SZ * D_SZ * sizeof(float)); // ~1.06 MB

  conv_silu_kernel<<<(B_SZ * L_SZ * D_SZ) / 256, 256, 0, stream>>>(
      x, conv_w, conv_b, xc);

  // 512 m-tiles, 4 waves (m-tiles) per 128-thread block -> 128 blocks
  proj_wmma_kernel<<<128, 128, 0, stream>>>(xc, xproj, proj);

  // 4096 channels, 16 channels per 256-thread block -> 256 blocks
  scan_kernel<<<(B_SZ * D_SZ) / 16, 256, 0, stream>>>(
      x, xc, proj, dt_w, dt_b, a_log, Dp, out);
}